// ADAG_72438918414732
// MI455X (gfx1250) — compile-verified
//
#include <hip/hip_runtime.h>
#include <hip/hip_bf16.h>
#include <math.h>

// Problem constants (match reference)
constexpr int kV = 500000;
constexpr int kB = 64;
constexpr int kN = 512;
constexpr int kE = 8192;
constexpr int kH = 128;
constexpr int kBN = kB * kN;     // 32768 rows

typedef float v2f __attribute__((ext_vector_type(2)));
typedef float v8f __attribute__((ext_vector_type(8)));

// ---------------------------------------------------------------------------
// GEMM: C[M,128] = act(A[M,K] @ W[K,128] (+ bias))   using V_WMMA_F32_16X16X4_F32
//
// Block: 256 threads = 8 waves (wave32). Tile: M=128 rows/block, N=128 (full),
// K staged through LDS in chunks of 32.
//  - A staged row-major [128][36] (pad -> fragment reads hit 16 distinct banks;
//    (k,k+1) pair is contiguous -> A fragment is one ds_load_b64).
//  - W staged k-pair interleaved: Wp[p][c] = {W[2p][c], W[2p+1][c]} (float2),
//    row stride 144 float2 (=288 floats == 32 mod 64 banks -> the two lane
//    halves use disjoint bank ranges). B fragment is one ds_load_b64, no
//    repacking movs.
// All staging uses float4 global loads + 16B-aligned vector LDS stores.
// Requires: M % 128 == 0, K % 32 == 0 (true here: M=32768, K in {256,128}).
// ---------------------------------------------------------------------------
#define AS_STRIDE 36     // floats per A row in LDS
#define WP_STRIDE 144    // float2 per Wp row in LDS

__global__ __launch_bounds__(256) void gemm_wmma_f32(
    const float* __restrict__ A, const float* __restrict__ W,
    const float* __restrict__ bias, float* __restrict__ C,
    int M, int K, int addBias, int relu)
{
  __shared__ float  As[128 * AS_STRIDE];   // 18.0 KB
  __shared__ float2 Wp[16 * WP_STRIDE];    // 18.0 KB

  const int tid  = threadIdx.x;
  const int wave = tid >> 5;
  const int lane = tid & 31;
  const int half = lane >> 4;      // 0: lanes 0-15, 1: lanes 16-31
  const int mr   = lane & 15;
  const int m0   = blockIdx.x * 128;
  const int wrow = m0 + wave * 16;

  v8f acc[8];
#pragma unroll
  for (int i = 0; i < 8; ++i)
#pragma unroll
    for (int j = 0; j < 8; ++j) acc[i][j] = 0.0f;

  const float* aRowBase = &As[(wave * 16 + mr) * AS_STRIDE + 2 * half];
  const float2* bColBase = &Wp[half * WP_STRIDE + (lane & 15)];

  for (int k0 = 0; k0 < K; k0 += 32) {
    // ---- stage A[128][32] chunk: 1024 float4s, 4 per thread, coalesced ----
#pragma unroll
    for (int i = 0; i < 4; ++i) {
      int idx = tid + i * 256;           // 0..1023
      int r = idx >> 3, c4 = idx & 7;    // row, float4-column
      float4 v = *(const float4*)&A[(size_t)(m0 + r) * K + k0 + c4 * 4];
      *(float4*)&As[r * AS_STRIDE + c4 * 4] = v;
    }
    // ---- stage W[32][128] chunk as 16 interleaved pair-rows ----
#pragma unroll
    for (int i = 0; i < 2; ++i) {
      int idx = tid + i * 256;           // 0..511
      int p = idx >> 5, g = idx & 31;    // pair-row, float4-group
      float4 a = *(const float4*)&W[(size_t)(k0 + 2 * p + 0) * kH + g * 4];
      float4 b = *(const float4*)&W[(size_t)(k0 + 2 * p + 1) * kH + g * 4];
      float4* dstp = (float4*)&Wp[p * WP_STRIDE + g * 4];
      dstp[0] = make_float4(a.x, b.x, a.y, b.y);
      dstp[1] = make_float4(a.z, b.z, a.w, b.w);
    }
    __syncthreads();

#pragma unroll
    for (int kk = 0; kk < 32; kk += 4) {
      // A fragment (16x4): lanes 0-15 hold K=kk,kk+1 ; lanes 16-31 K=kk+2,kk+3
      const float2 av = *(const float2*)(aRowBase + kk);
      v2f afrag; afrag.x = av.x; afrag.y = av.y;
      const float2* bRow = bColBase + (kk >> 1) * WP_STRIDE;
#pragma unroll
      for (int n = 0; n < 8; ++n) {
        float2 bv = bRow[n * 16];
        v2f bfrag; bfrag.x = bv.x; bfrag.y = bv.y;
        acc[n] = __builtin_amdgcn_wmma_f32_16x16x4_f32(
            false, afrag, false, bfrag, (short)0, acc[n], false, false);
      }
    }
    __syncthreads();
  }

  // C 16x16 f32 layout: VGPR v -> row v (lanes 0-15) / row v+8 (lanes 16-31)
#pragma unroll
  for (int n = 0; n < 8; ++n) {
#pragma unroll
    for (int v = 0; v < 8; ++v) {
      int row = wrow + v + 8 * half;
      int col = n * 16 + mr;
      float val = acc[n][v];
      if (addBias) val += bias[col];
      if (relu) val = fmaxf(val, 0.0f);
      C[(size_t)row * kH + col] = val;
    }
  }
}

// ---------------------------------------------------------------------------
// Per-graph degree -> dinv = rsqrt(in_degree + 1)   (self loop included)
// ---------------------------------------------------------------------------
__global__ __launch_bounds__(256) void compute_dinv_kernel(
    const int* __restrict__ edges, float* __restrict__ dinv)
{
  __shared__ int deg[kN];
  const int b = blockIdx.x;
  const int tid = threadIdx.x;
  for (int n = tid; n < kN; n += 256) deg[n] = 1;   // self loop
  __syncthreads();
  const int* dst = edges + (size_t)b * 2 * kE + kE; // edge[1]
  for (int e = tid; e < kE; e += 256) atomicAdd(&deg[dst[e]], 1);
  __syncthreads();
  for (int n = tid; n < kN; n += 256)
    dinv[b * kN + n] = rsqrtf((float)deg[n]);
}

// ---------------------------------------------------------------------------
// GCN scatter-aggregation with the full 512x128 fp32 tile resident in LDS
// (256KB dynamic LDS -- exploits CDNA5's 320KB/WGP). 512 threads = 16 waves
// issuing ds_add_f32 atomics in parallel (LDS caps us at 1 WG/WGP anyway).
// out[b,d,:] = sum_{e: s->d} h[b,s,:]*dinv[s]*dinv[d] + h[b,d,:]*dinv[d]^2
//              + bias ; optional ReLU
// ---------------------------------------------------------------------------
__global__ __launch_bounds__(512) void gcn_aggregate_kernel(
    const float* __restrict__ Hm, const int* __restrict__ edges,
    const float* __restrict__ dinv, const float* __restrict__ bias,
    float* __restrict__ out, int relu)
{
  extern __shared__ float acc[];                  // kN*kH floats = 256 KB
  const int b   = blockIdx.x;
  const int tid = threadIdx.x;
  const int wave = tid >> 5, lane = tid & 31;
  const float* hb = Hm + (size_t)b * kN * kH;
  const float* db = dinv + b * kN;

  float4* accv = (float4*)acc;
  for (int i = tid; i < kN * kH / 4; i += 512)
    accv[i] = make_float4(0.f, 0.f, 0.f, 0.f);
  __syncthreads();

  const int* src = edges + (size_t)b * 2 * kE;
  const int* dst = src + kE;

  // real edges: one wave per edge, each lane owns a float4 slice
  for (int e = wave; e < kE; e += 16) {
    int s = src[e], d = dst[e];
    float nm = db[s] * db[d];
    float4 v = ((const float4*)(hb + (size_t)s * kH))[lane];
    float* p = &acc[(size_t)d * kH + lane * 4];
    atomicAdd(p + 0, v.x * nm);
    atomicAdd(p + 1, v.y * nm);
    atomicAdd(p + 2, v.z * nm);
    atomicAdd(p + 3, v.w * nm);
  }
  // self loops
  for (int n = wave; n < kN; n += 16) {
    float dn = db[n];
    float nm = dn * dn;
    float4 v = ((const float4*)(hb + (size_t)n * kH))[lane];
    float* p = &acc[(size_t)n * kH + lane * 4];
    atomicAdd(p + 0, v.x * nm);
    atomicAdd(p + 1, v.y * nm);
    atomicAdd(p + 2, v.z * nm);
    atomicAdd(p + 3, v.w * nm);
  }
  __syncthreads();

  const float4* bv = (const float4*)bias;         // 32 float4s
  float4* ob = (float4*)(out + (size_t)b * kN * kH);
  for (int i = tid; i < kN * kH / 4; i += 512) {
    float4 a = accv[i];
    float4 bb = bv[i & 31];
    a.x += bb.x; a.y += bb.y; a.z += bb.z; a.w += bb.w;
    if (relu) {
      a.x = fmaxf(a.x, 0.f); a.y = fmaxf(a.y, 0.f);
      a.z = fmaxf(a.z, 0.f); a.w = fmaxf(a.w, 0.f);
    }
    ob[i] = a;
  }
}

// ---------------------------------------------------------------------------
// Gathers (float4 rows; 64 threads per 256-float output row)
// ---------------------------------------------------------------------------
__global__ __launch_bounds__(256) void gather_pos_kernel(
    const int* __restrict__ node_ids, const float* __restrict__ emb,
    const float* __restrict__ feat, float* __restrict__ xin)
{
  int row = blockIdx.x * 4 + (threadIdx.x >> 6);  // [0, kBN)
  int t = threadIdx.x & 63;
  int id = node_ids[row];
  float4* dstv = (float4*)xin + (size_t)row * 64;
  if (t < 32)
    dstv[t] = ((const float4*)emb)[(size_t)id * 32 + t];
  else
    dstv[t] = ((const float4*)feat)[(size_t)id * 32 + (t - 32)];
}

__global__ __launch_bounds__(256) void gather_mal_kernel(
    const int* __restrict__ mal_posidx, const int* __restrict__ mal_nodes,
    const float* __restrict__ pos_emb, const float* __restrict__ feat,
    float* __restrict__ xin)
{
  int row = blockIdx.x * 4 + (threadIdx.x >> 6);
  int t = threadIdx.x & 63;
  int b = row >> 9;                 // row / kN
  float4* dstv = (float4*)xin + (size_t)row * 64;
  if (t < 32) {
    int p = mal_posidx[row];
    dstv[t] = ((const float4*)pos_emb)[((size_t)b * kN + p) * 32 + t];
  } else {
    int id = mal_nodes[row];
    dstv[t] = ((const float4*)feat)[(size_t)id * 32 + (t - 32)];
  }
}

__global__ __launch_bounds__(128) void gather_root_kernel(
    const float* __restrict__ src, const int* __restrict__ root_idx,
    float* __restrict__ dstp)
{
  int b = blockIdx.x, t = threadIdx.x;
  dstp[b * kH + t] = src[((size_t)b * kN + root_idx[b]) * kH + t];
}

// ---------------------------------------------------------------------------
// score(r) = sigmoid( relu(r @ W1 + b1) @ W2 + b2 )   -- one block per graph
// ---------------------------------------------------------------------------
__global__ __launch_bounds__(128) void score_kernel(
    const float* __restrict__ r, const float* __restrict__ W1,
    const float* __restrict__ b1, const float* __restrict__ W2,
    const float* __restrict__ b2, float* __restrict__ out)
{
  __shared__ float rs[kH];
  __shared__ float red[kH];
  int b = blockIdx.x, j = threadIdx.x;
  rs[j] = r[b * kH + j];
  __syncthreads();
  float a = b1[j];
  for (int i = 0; i < kH; ++i) a += rs[i] * W1[i * kH + j];
  red[j] = fmaxf(a, 0.f) * W2[j];
  __syncthreads();
  for (int s = 64; s > 0; s >>= 1) {
    if (j < s) red[j] += red[j + s];
    __syncthreads();
  }
  if (j == 0) out[b] = 1.f / (1.f + expf(-(red[0] + b2[0])));
}

// ---------------------------------------------------------------------------
// Host orchestration
// ---------------------------------------------------------------------------
extern "C" void kernel_launch(void* const* d_in, const int* in_sizes, int n_in,
                              void* d_out, int out_size, void* d_ws, size_t ws_size,
                              hipStream_t stream) {
  const int*   node_ids    = (const int*)d_in[0];
  const int*   edge_pos    = (const int*)d_in[1];
  const int*   root_idx    = (const int*)d_in[2];
  const int*   mal_nodes1  = (const int*)d_in[3];
  const int*   mal_posidx1 = (const int*)d_in[4];
  const int*   edge_m1     = (const int*)d_in[5];
  const int*   mal_nodes2  = (const int*)d_in[6];
  const int*   mal_posidx2 = (const int*)d_in[7];
  const int*   edge_m2     = (const int*)d_in[8];
  const float* embeddings  = (const float*)d_in[9];
  const float* features    = (const float*)d_in[10];
  const float* fe_W1 = (const float*)d_in[11];
  const float* fe_b1 = (const float*)d_in[12];
  const float* fe_W2 = (const float*)d_in[13];
  const float* fe_b2 = (const float*)d_in[14];
  const float* g1_W  = (const float*)d_in[15];
  const float* g1_b  = (const float*)d_in[16];
  const float* g2_W  = (const float*)d_in[17];
  const float* g2_b  = (const float*)d_in[18];
  const float* mlp_W1  = (const float*)d_in[19];
  const float* mlp_b1  = (const float*)d_in[20];
  const float* mlp_W2  = (const float*)d_in[21];
  const float* mlp_b2  = (const float*)d_in[22];
  const float* mlp1_W1 = (const float*)d_in[23];
  const float* mlp1_b1 = (const float*)d_in[24];
  const float* mlp1_W2 = (const float*)d_in[25];
  const float* mlp1_b2 = (const float*)d_in[26];
  float* out = (float*)d_out;

  // Workspace carve-up (floats)
  float* ws   = (float*)d_ws;
  float* xin  = ws;                              // kBN*256
  float* xA   = xin  + (size_t)kBN * 256;        // kBN*128
  float* xB   = xA   + (size_t)kBN * kH;         // kBN*128
  float* xC   = xB   + (size_t)kBN * kH;         // kBN*128 (mal branch output)
  float* pos  = xC   + (size_t)kBN * kH;         // kBN*128 (pos_emb, persists)
  float* dinv = pos  + (size_t)kBN * kH;         // kBN
  float* roots = dinv + (size_t)kBN;             // 3*kB*kH

  // Allow 256KB dynamic LDS for the aggregation kernel (host-side, capture-safe)
  (void)hipFuncSetAttribute((const void*)gcn_aggregate_kernel,
                            hipFuncAttributeMaxDynamicSharedMemorySize,
                            kN * kH * (int)sizeof(float));

  const dim3 gemmGrid(kBN / 128);
  const size_t aggLds = (size_t)kN * kH * sizeof(float);

  auto run_branch = [&](const int* edges, float* outbuf) {
    // fe_mlp
    gemm_wmma_f32<<<gemmGrid, 256, 0, stream>>>(xin, fe_W1, fe_b1, xA, kBN, 256, 1, 1);
    gemm_wmma_f32<<<gemmGrid, 256, 0, stream>>>(xA,  fe_W2, fe_b2, xB, kBN, kH, 1, 0);
    // degrees for this branch's edges
    compute_dinv_kernel<<<kB, 256, 0, stream>>>(edges, dinv);
    // gcn1: h = x @ W ; aggregate + bias + relu
    gemm_wmma_f32<<<gemmGrid, 256, 0, stream>>>(xB, g1_W, nullptr, xA, kBN, kH, 0, 0);
    gcn_aggregate_kernel<<<kB, 512, aggLds, stream>>>(xA, edges, dinv, g1_b, xB, 1);
    // gcn2
    gemm_wmma_f32<<<gemmGrid, 256, 0, stream>>>(xB, g2_W, nullptr, xA, kBN, kH, 0, 0);
    gcn_aggregate_kernel<<<kB, 512, aggLds, stream>>>(xA, edges, dinv, g2_b, outbuf, 0);
  };

  // ---- positive branch ----
  gather_pos_kernel<<<kBN / 4, 256, 0, stream>>>(node_ids, embeddings, features, xin);
  run_branch(edge_pos, pos);
  gather_root_kernel<<<kB, kH, 0, stream>>>(pos, root_idx, roots + 0 * kB * kH);

  // ---- malicious branch 1 ----
  gather_mal_kernel<<<kBN / 4, 256, 0, stream>>>(mal_posidx1, mal_nodes1, pos, features, xin);
  run_branch(edge_m1, xC);
  gather_root_kernel<<<kB, kH, 0, stream>>>(xC, root_idx, roots + 1 * kB * kH);

  // ---- malicious branch 2 ----
  gather_mal_kernel<<<kBN / 4, 256, 0, stream>>>(mal_posidx2, mal_nodes2, pos, features, xin);
  run_branch(edge_m2, xC);
  gather_root_kernel<<<kB, kH, 0, stream>>>(xC, root_idx, roots + 2 * kB * kH);

  // ---- scores (output order: pos, pos1, mal1, mal2) ----
  score_kernel<<<kB, kH, 0, stream>>>(roots + 0 * kB * kH, mlp_W1,  mlp_b1,  mlp_W2,  mlp_b2,  out + 0 * kB);
  score_kernel<<<kB, kH, 0, stream>>>(roots + 0 * kB * kH, mlp1_W1, mlp1_b1, mlp1_W2, mlp1_b2, out + 1 * kB);
  score_kernel<<<kB, kH, 0, stream>>>(roots + 1 * kB * kH, mlp_W1,  mlp_b1,  mlp_W2,  mlp_b2,  out + 2 * kB);
  score_kernel<<<kB, kH, 0, stream>>>(roots + 2 * kB * kH, mlp1_W1, mlp1_b1, mlp1_W2, mlp1_b2, out + 3 * kB);
}